// NTXentLoss_44246753084252
// MI455X (gfx1250) — compile-verified
//
#include <hip/hip_runtime.h>
#include <math.h>

// NT-Xent loss, fused: sim tiles computed with V_WMMA_F32_16X16X4_F32 (exact
// f32 matrix math), exp+rowsum folded in, never materializing the 8192x8192
// similarity matrix. Normalization folded into per-tile scaling by inverse
// row norms. Block mapping: the 4 waves of a block own 4 adjacent 16-row A
// strips (A register-resident) and sweep the SAME column-tile sequence, so
// their streamed B loads are identical addresses -> L0 reuse x4.

typedef __attribute__((ext_vector_type(2))) float v2f;
typedef __attribute__((ext_vector_type(8))) float v8f;

static constexpr int kD   = 256;   // feature dim
static constexpr int kBh  = 4096;  // half batch B
static constexpr int kN   = 8192;  // 2B rows
static constexpr float kTempInv = 2.0f;   // 1 / 0.5
static constexpr float kEps     = 1e-8f;

// ---------------------------------------------------------------------------
// Kernel 1: rnorm[r] = 1 / max(||z_r||, eps); zero denom/pos accumulators.
// One block (256 threads = 8 waves) per row.
// ---------------------------------------------------------------------------
__global__ __launch_bounds__(256)
void ntx_rownorm(const float* __restrict__ zi, const float* __restrict__ zj,
                 float* __restrict__ rnorm, float* __restrict__ denom,
                 float* __restrict__ pos) {
  const int row = blockIdx.x;
  const float* src = (row < kBh) ? (zi + (size_t)row * kD)
                                 : (zj + (size_t)(row - kBh) * kD);
  float x = src[threadIdx.x];
  float s = x * x;
  #pragma unroll
  for (int m = 16; m >= 1; m >>= 1) s += __shfl_xor(s, m, 32);

  __shared__ float ls[8];
  const int wave = threadIdx.x >> 5;
  const int lane = threadIdx.x & 31;
  if (lane == 0) ls[wave] = s;
  __syncthreads();
  if (threadIdx.x == 0) {
    float t = 0.0f;
    #pragma unroll
    for (int i = 0; i < 8; ++i) t += ls[i];
    rnorm[row] = 1.0f / fmaxf(sqrtf(t), kEps);
    denom[row] = 0.0f;
    pos[row]   = 0.0f;
  }
}

// ---------------------------------------------------------------------------
// Kernel 2: fused sim = (z zT) * rn_r * rn_c / T, exp, row-sum, positive
// extraction. grid = (row blocks = 128 of 64 rows, column quarters = 4),
// 128 threads = 4 waves; wave w owns row tile ic = (bx*4+w)*16 and sweeps
// the quarter's 128 column tiles, 64 x v_wmma_f32_16x16x4_f32 each.
//
// A (16x4 f32) lane layout: lanes 0-15 hold A[l][k0,k0+1], lanes 16-31 hold
// A[l-16][k0+2,k0+3]  -> float2 load at z[ic+l%16][k0 + 2*(l>=16)].
// B (4x16 f32) lane layout mirrors it with the column base jc.
// C/D: lanes 0-15 -> M=v (VGPR v), N=lane; lanes 16-31 -> M=8+v, N=lane-16.
// ---------------------------------------------------------------------------
__global__ __launch_bounds__(128)
void ntx_sim(const float* __restrict__ zi, const float* __restrict__ zj,
             const float* __restrict__ rnorm, float* __restrict__ denom,
             float* __restrict__ pos) {
  const int lane = threadIdx.x & 31;
  const int wave = threadIdx.x >> 5;
  const int l15  = lane & 15;
  const int hi   = lane >> 4;           // 0: K pair {0,1} / M 0-7, 1: {2,3} / M 8-15

  const int ic   = (blockIdx.x * 4 + wave) * 16;   // this wave's row-tile base
  const int arow = ic + l15;
  const float* aptr = ((arow < kBh) ? (zi + (size_t)arow * kD)
                                    : (zj + (size_t)(arow - kBh) * kD)) + 2 * hi;

  // inverse norms of the 8 accumulator rows owned by this half-wave (uniform)
  float rnR[8];
  #pragma unroll
  for (int v = 0; v < 8; ++v) rnR[v] = rnorm[ic + v + 8 * hi];

  float rsum[8];
  float psum[8];
  #pragma unroll
  for (int v = 0; v < 8; ++v) { rsum[v] = 0.0f; psum[v] = 0.0f; }

  // All 4 waves of the block sweep the same column tiles (B loads shared in L0).
  const int tilesPerQuarter = (kN / 16) / 4;              // 128 column tiles
  const int jtEnd = (blockIdx.y + 1) * tilesPerQuarter;
  for (int jt = blockIdx.y * tilesPerQuarter; jt < jtEnd; ++jt) {
    const int jc   = jt * 16;
    const int brow = jc + l15;                            // this lane's column
    const float* bptr = ((brow < kBh) ? (zi + (size_t)brow * kD)
                                      : (zj + (size_t)(brow - kBh) * kD)) + 2 * hi;
    const float rnC = rnorm[brow];

    v8f acc = {};
    #pragma unroll
    for (int k = 0; k < kD; k += 4) {
      v2f a = *(const v2f*)(aptr + k);
      v2f b = *(const v2f*)(bptr + k);
      acc = __builtin_amdgcn_wmma_f32_16x16x4_f32(
          /*neg_a=*/false, a, /*neg_b=*/false, b,
          /*c_mod=*/(short)0, acc, /*reuse_a=*/false, /*reuse_b=*/false);
    }

    #pragma unroll
    for (int v = 0; v < 8; ++v) {
      const int r = ic + v + 8 * hi;                      // global row
      const int c = brow;                                 // global column
      const float sim = acc[v] * (rnR[v] * rnC * kTempInv);
      const float e = (c == r) ? 0.0f : __expf(sim);      // diag -> exp(-inf)=0
      rsum[v] += e;
      if ((c == r + kBh) || (c + kBh == r)) psum[v] += sim; // the one positive
    }
  }

  // reduce the 16 column-lanes of each half-wave; lanes 0 and 16 commit.
  #pragma unroll
  for (int v = 0; v < 8; ++v) {
    float r_ = rsum[v];
    float p_ = psum[v];
    #pragma unroll
    for (int m = 8; m >= 1; m >>= 1) {
      r_ += __shfl_xor(r_, m, 32);
      p_ += __shfl_xor(p_, m, 32);
    }
    if (l15 == 0) {
      const int r = ic + v + 8 * hi;
      atomicAdd(&denom[r], r_);
      atomicAdd(&pos[r], p_);
    }
  }
}

// ---------------------------------------------------------------------------
// Kernel 3: loss = mean_i( log(denom_i) - pos_i )
// ---------------------------------------------------------------------------
__global__ __launch_bounds__(256)
void ntx_finalize(const float* __restrict__ denom, const float* __restrict__ pos,
                  float* __restrict__ out) {
  float s = 0.0f;
  for (int i = threadIdx.x; i < kN; i += 256)
    s += logf(denom[i]) - pos[i];
  #pragma unroll
  for (int m = 16; m >= 1; m >>= 1) s += __shfl_xor(s, m, 32);

  __shared__ float ls[8];
  const int wave = threadIdx.x >> 5;
  const int lane = threadIdx.x & 31;
  if (lane == 0) ls[wave] = s;
  __syncthreads();
  if (threadIdx.x == 0) {
    float t = 0.0f;
    #pragma unroll
    for (int i = 0; i < 8; ++i) t += ls[i];
    out[0] = t / (float)kN;
  }
}

extern "C" void kernel_launch(void* const* d_in, const int* in_sizes, int n_in,
                              void* d_out, int out_size, void* d_ws, size_t ws_size,
                              hipStream_t stream) {
  const float* zi = (const float*)d_in[0];   // [4096, 256] f32
  const float* zj = (const float*)d_in[1];   // [4096, 256] f32
  float* out = (float*)d_out;                // scalar loss

  float* rnorm = (float*)d_ws;               // [8192]
  float* denom = rnorm + kN;                 // [8192]
  float* pos   = denom + kN;                 // [8192]

  ntx_rownorm<<<kN, 256, 0, stream>>>(zi, zj, rnorm, denom, pos);
  ntx_sim<<<dim3(kN / 64, 4), 128, 0, stream>>>(zi, zj, rnorm, denom, pos);
  ntx_finalize<<<1, 256, 0, stream>>>(denom, pos, out);
}